// GaussianMomentDescriptor_8211977470391
// MI455X (gfx1250) — compile-verified
//
#include <hip/hip_runtime.h>
#include <math.h>

// ---------------------------------------------------------------------------
// GaussianMomentDescriptor for MI455X (gfx1250, wave32, WMMA).
//
// Phase 1: zero 4MB compact-moment workspace (L2 resident).
// Phase 2: edge kernel — per-edge radial basis + symmetric-compressed moment
//          scatter: 100 global_atomic_add_f32 per edge (vs 200 naive).
// Phase 3: atom kernel — one wave per atom; contractions c1..c7 as small GEMMs
//          on V_WMMA_F32_16X16X4_F32. Operands are zero-padded in LDS so every
//          fragment load is an unconditional aligned ds_load_b64 (no EXEC
//          diamonds), and EXEC stays all-ones across each WMMA.
// ---------------------------------------------------------------------------

#define NR    5          // n_radial
#define NB    7          // n_basis
#define CMPT  100        // compact floats/atom: 5 m0 + 15 m1 + 30 m2u + 50 m3u
#define OUTD  360

typedef float v2f __attribute__((ext_vector_type(2)));
typedef float v8f __attribute__((ext_vector_type(8)));

// tril3 indices over 5 radial (k<=j<=i): 35 triples (one-shot extraction only)
__device__ const int kI3[35] = {0, 1,1,1, 2,2,2,2,2,2, 3,3,3,3,3,3,3,3,3,3,
                                4,4,4,4,4,4,4,4,4,4,4,4,4,4,4};
__device__ const int kJ3[35] = {0, 0,1,1, 0,1,1,2,2,2, 0,1,1,2,2,2,3,3,3,3,
                                0,1,1,2,2,2,3,3,3,3,4,4,4,4,4};
__device__ const int kK3[35] = {0, 0,0,1, 0,0,1,0,1,2, 0,0,1,0,1,2,0,1,2,3,
                                0,0,1,0,1,2,0,1,2,3,0,1,2,3,4};
// full (i,j,k) -> compact symmetric-3 index (sorted-triple order, 10 entries)
__device__ const int kP3[27] = {0,1,2, 1,3,4, 2,4,5,
                                1,3,4, 3,6,7, 4,7,8,
                                2,4,5, 4,7,8, 5,8,9};

// branch-free tril2 inverse over 5: p in [0,15) -> (i,j), j<=i
__device__ __forceinline__ void tril2_inv(int p, int& i, int& j) {
  i = (p >= 1) + (p >= 3) + (p >= 6) + (p >= 10);
  j = p - (i * (i + 1)) / 2;
}
// (x,y) -> compact symmetric-2 index over 3 (order 00,01,02,11,12,22)
__device__ __forceinline__ int sym2_idx(int x, int y) {
  int a = min(x, y), b = max(x, y);
  return (a * (5 - a)) / 2 + b;
}

__device__ __forceinline__ void atomAddF(float* p, float v) {
  // forces hw global_atomic_add_f32 (no-return), avoids CAS loop
  (void)unsafeAtomicAdd(p, v);
}

// ---------------------------------------------------------------------------
__global__ void gmd_zero_kernel(float* __restrict__ p, int n) {
  int i = blockIdx.x * blockDim.x + threadIdx.x;
  if (i < n) p[i] = 0.0f;
}

// ---------------------------------------------------------------------------
// Edge scatter kernel: one thread per edge.
__global__ void gmd_edge_kernel(const float* __restrict__ R,
                                const int*   __restrict__ Z,
                                const int*   __restrict__ nbr,
                                const float* __restrict__ emb,
                                float*       __restrict__ mom,
                                int n_edges, int n_species) {
  int e = blockIdx.x * blockDim.x + threadIdx.x;
  if (e >= n_edges) return;

  const int i = nbr[e];
  const int j = nbr[n_edges + e];

  float dx = R[3*j+0] - R[3*i+0];
  float dy = R[3*j+1] - R[3*i+1];
  float dz = R[3*j+2] - R[3*i+2];
  float dr = sqrtf(dx*dx + dy*dy + dz*dz);
  float inv = 1.0f / (dr + 1e-5f);
  float dnv[3] = {dx*inv, dy*inv, dz*inv};

  // Gaussian basis: betta = 49/36, rad_norm = (2*betta/pi)^0.25
  const float betta    = 1.36111111f;
  const float rad_norm = 0.96482115f;
  const float dshift   = 0.785714286f;      // (6.0-0.5)/7
  float basis[NB];
#pragma unroll
  for (int b = 0; b < NB; ++b) {
    float t = (0.5f + dshift * (float)b) - dr;
    basis[b] = rad_norm * __expf(-betta * t * t);
  }

  // species-pair embedding contraction: radial[r] = (1/sqrt(7)) * C[r,:].basis
  const float inv_sqrt_nb = 0.377964473f;
  const float* cb = emb + ((size_t)Z[j] * (size_t)n_species + (size_t)Z[i]) * (NR * NB);
  float radial[NR];
#pragma unroll
  for (int r = 0; r < NR; ++r) {
    float acc = 0.0f;
#pragma unroll
    for (int b = 0; b < NB; ++b) acc += cb[r*NB + b] * basis[b];
    radial[r] = acc * inv_sqrt_nb;
  }

  // cosine cutoff + self-edge mask
  float dc  = fminf(dr, 6.0f);
  float cut = 0.5f * (__cosf(3.14159265358979f * dc / 6.0f) + 1.0f);
  if (i == j) cut = 0.0f;

  // unique direction products (symmetric compression)
  float dd2[6], dd3[10];
  {
    int p = 0;
#pragma unroll
    for (int x = 0; x < 3; ++x)
#pragma unroll
      for (int y = x; y < 3; ++y) dd2[p++] = dnv[x] * dnv[y];
    p = 0;
#pragma unroll
    for (int x = 0; x < 3; ++x)
#pragma unroll
      for (int y = x; y < 3; ++y)
#pragma unroll
        for (int z = y; z < 3; ++z) dd3[p++] = dnv[x] * dnv[y] * dnv[z];
  }

  float* base = mom + (size_t)j * CMPT;
#pragma unroll
  for (int r = 0; r < NR; ++r) {
    float s = radial[r] * cut;
    atomAddF(base + r, s);
#pragma unroll
    for (int x = 0; x < 3; ++x)  atomAddF(base + 5  + r*3  + x, s * dnv[x]);
#pragma unroll
    for (int p = 0; p < 6; ++p)  atomAddF(base + 20 + r*6  + p, s * dd2[p]);
#pragma unroll
    for (int p = 0; p < 10; ++p) atomAddF(base + 50 + r*10 + p, s * dd3[p]);
  }
}

// ---------------------------------------------------------------------------
// WMMA helper: D(16x16) = A(16xK) * Bsrc^T using V_WMMA_F32_16X16X4_F32.
// A/Bsrc are LDS, row-major, K padded to a multiple of 4 and rows zero-padded,
// so all fragment loads are unconditional aligned 8-byte LDS loads.
// f32 16x4 A layout (ISA 7.12.2): lanes 0-15 hold K={c,c+1}, lanes 16-31 {c+2,c+3}.
template <int SA, int SB, int K>
__device__ __forceinline__ v8f gemmT(const float* __restrict__ A,
                                     const float* __restrict__ B, int lane) {
  v8f acc = {0.f,0.f,0.f,0.f,0.f,0.f,0.f,0.f};
  const int idx = lane & 15;
  const int ko  = (lane >> 4) * 2;
#pragma unroll
  for (int c = 0; c < K; c += 4) {
    v2f a = *(const v2f*)(A + idx * SA + c + ko);
    v2f b = *(const v2f*)(B + idx * SB + c + ko);
    acc = __builtin_amdgcn_wmma_f32_16x16x4_f32(false, a, false, b,
                                                (short)0, acc, false, false);
  }
  return acc;
}

// D fragment -> LDS [row*16+col]; lane holds col=lane&15 of rows v + 8*(lane>>4)
__device__ __forceinline__ void dumpD(v8f d, float* __restrict__ sD, int lane) {
  const int half = lane >> 4;
  const int idx  = lane & 15;
#pragma unroll
  for (int v = 0; v < 8; ++v) sD[(v + half * 8) * 16 + idx] = d[v];
}

// ---------------------------------------------------------------------------
// Atom kernel: one wave (32 threads) per atom.
// LDS pool layout (floats), all operand regions zero-padded:
//   sM1 [16 x 4]   @ 0       sM2 [16 x 12] @ 64     sM3 [16 x 28] @ 256
//   sA  [32 x 12]  @ 704     sD  [16 x 16] @ 1088   sStage [100]  @ 1344
#define POOL_SZ 1444
__global__ __launch_bounds__(32)
void gmd_atom_kernel(const float* __restrict__ mom, float* __restrict__ out) {
  __shared__ float sPool[POOL_SZ];
  float* sM1    = sPool;
  float* sM2    = sPool + 64;
  float* sM3    = sPool + 256;
  float* sA     = sPool + 704;
  float* sD     = sPool + 1088;
  float* sStage = sPool + 1344;

  const int a    = blockIdx.x;
  const int lane = threadIdx.x;
  const float* mb = mom + (size_t)a * CMPT;
  float*       ob = out + (size_t)a * OUTD;

  for (int t = lane; t < 1088; t += 32) sPool[t] = 0.0f;   // zero operand pads
  for (int t = lane; t < CMPT; t += 32) sStage[t] = mb[t];
  __syncthreads();

  if (lane < NR) ob[lane] = sStage[lane];                  // m0
  for (int t = lane; t < NR * 3; t += 32)                  // m1 -> [16x4]
    sM1[(t / 3) * 4 + (t % 3)] = sStage[5 + t];
  for (int t = lane; t < NR * 9; t += 32) {                // expand m2 -> [16x12]
    int r = t / 9, xy = t % 9;
    sM2[r * 12 + xy] = sStage[20 + r * 6 + sym2_idx(xy / 3, xy % 3)];
  }
  for (int t = lane; t < NR * 27; t += 32) {               // expand m3 -> [16x28]
    int r = t / 27, w = t % 27;
    sM3[r * 28 + w] = sStage[50 + r * 10 + kP3[w]];
  }
  __syncthreads();

  // ---- c1 = M1*M1^T (K=3), c2 = M2*M2^T (K=9), c3 = M3*M3^T (K=27) ----
  v8f d = gemmT<4, 4, 4>(sM1, sM1, lane);
  dumpD(d, sD, lane); __syncthreads();
  for (int p = lane; p < 15; p += 32) {
    int i, j; tril2_inv(p, i, j);
    ob[5 + p] = sD[i * 16 + j];
  }
  __syncthreads();

  d = gemmT<12, 12, 12>(sM2, sM2, lane);
  dumpD(d, sD, lane); __syncthreads();
  for (int p = lane; p < 15; p += 32) {
    int i, j; tril2_inv(p, i, j);
    ob[20 + p] = sD[i * 16 + j];
  }
  __syncthreads();

  d = gemmT<28, 28, 28>(sM3, sM3, lane);
  dumpD(d, sD, lane); __syncthreads();
  for (int p = lane; p < 15; p += 32) {
    int i, j; tril2_inv(p, i, j);
    ob[35 + p] = sD[i * 16 + j];
  }
  __syncthreads();

  // ---- c4: Q[p=(r,s)][j*3+k] = sum_i m2[r,ij] m2[s,ik]; c4 = Q * M2^T ----
  for (int t = lane; t < 15 * 9; t += 32) {
    int p = t / 9, jk = t % 9, jj = jk / 3, kk = jk % 3;
    int r, s; tril2_inv(p, r, s);
    float acc = 0.0f;
#pragma unroll
    for (int ii = 0; ii < 3; ++ii)
      acc += sM2[r * 12 + ii * 3 + jj] * sM2[s * 12 + ii * 3 + kk];
    sA[p * 12 + jk] = acc;
  }
  __syncthreads();
  d = gemmT<12, 12, 12>(sA, sM2, lane);
  dumpD(d, sD, lane); __syncthreads();
  for (int q = lane; q < 35; q += 32) {
    int i = kI3[q], jj = kJ3[q], kk = kK3[q];
    int p = i * (i + 1) / 2 + jj;                          // tril2 pair index
    ob[50 + q] = sD[p * 16 + kk];
  }
  __syncthreads();

  // ---- c5: P[p=(r,s)][i*3+j] = m1[r,i]*m1[s,j]; c5 = P * M2^T ----
  for (int t = lane; t < 15 * 9; t += 32) {
    int p = t / 9, ij = t % 9;
    int r, s; tril2_inv(p, r, s);
    sA[p * 12 + ij] = sM1[r * 4 + ij / 3] * sM1[s * 4 + ij % 3];
  }
  __syncthreads();
  d = gemmT<12, 12, 12>(sA, sM2, lane);
  dumpD(d, sD, lane); __syncthreads();
  for (int e2 = lane; e2 < 75; e2 += 32)
    ob[85 + e2] = sD[(e2 / 5) * 16 + (e2 % 5)];
  __syncthreads();

  // ---- c6: T[p=(r,s)][k*3+l] = sum_ij m3[r,(ij)k] m3[s,(ij)l]; c6 = T*M2^T --
  for (int t = lane; t < 15 * 9; t += 32) {
    int p = t / 9, kl = t % 9, kk = kl / 3, ll = kl % 3;
    int r, s; tril2_inv(p, r, s);
    float acc = 0.0f;
#pragma unroll
    for (int ij = 0; ij < 9; ++ij)
      acc += sM3[r * 28 + ij * 3 + kk] * sM3[s * 28 + ij * 3 + ll];
    sA[p * 12 + kl] = acc;
  }
  __syncthreads();
  d = gemmT<12, 12, 12>(sA, sM2, lane);
  dumpD(d, sD, lane); __syncthreads();
  for (int e2 = lane; e2 < 75; e2 += 32)
    ob[160 + e2] = sD[(e2 / 5) * 16 + (e2 % 5)];
  __syncthreads();

  // ---- c7: H[p=r*5+t][ij] = sum_k m3[r,(ij)k] m1[t,k]; c7 = H * M2^T ----
  for (int t = lane; t < 25 * 9; t += 32) {
    int p = t / 9, ij = t % 9;
    int r = p / 5, tt = p % 5;
    float acc = 0.0f;
#pragma unroll
    for (int kk = 0; kk < 3; ++kk)
      acc += sM3[r * 28 + ij * 3 + kk] * sM1[tt * 4 + kk];
    sA[p * 12 + ij] = acc;
  }
  __syncthreads();
  d = gemmT<12, 12, 12>(sA, sM2, lane);                    // pairs 0..15
  dumpD(d, sD, lane); __syncthreads();
  for (int e2 = lane; e2 < 80; e2 += 32) {
    int p = e2 / 5, s = e2 % 5, r = p / 5, tt = p % 5;
    ob[235 + r * 25 + s * 5 + tt] = sD[p * 16 + s];
  }
  __syncthreads();
  d = gemmT<12, 12, 12>(sA + 16 * 12, sM2, lane);          // pairs 16..24
  dumpD(d, sD, lane); __syncthreads();
  for (int e2 = lane; e2 < 45; e2 += 32) {
    int pr = e2 / 5, s = e2 % 5;
    int p = 16 + pr, r = p / 5, tt = p % 5;
    ob[235 + r * 25 + s * 5 + tt] = sD[pr * 16 + s];
  }
}

// ---------------------------------------------------------------------------
extern "C" void kernel_launch(void* const* d_in, const int* in_sizes, int n_in,
                              void* d_out, int out_size, void* d_ws, size_t ws_size,
                              hipStream_t stream) {
  const float* R   = (const float*)d_in[0];
  const int*   Z   = (const int*)d_in[1];
  const int*   nbr = (const int*)d_in[2];
  // d_in[3] = box, unused (free-space branch)
  const float* emb = (const float*)d_in[4];

  const int n_atoms = in_sizes[0] / 3;
  const int n_edges = in_sizes[2] / 2;
  const int sp2     = in_sizes[4] / (NR * NB);
  int n_species = (int)(sqrtf((float)sp2) + 0.5f);

  float* moments = (float*)d_ws;                 // n_atoms * 100 floats (4 MB)

  const int nz = n_atoms * CMPT;
  gmd_zero_kernel<<<(nz + 255) / 256, 256, 0, stream>>>(moments, nz);
  gmd_edge_kernel<<<(n_edges + 255) / 256, 256, 0, stream>>>(
      R, Z, nbr, emb, moments, n_edges, n_species);
  gmd_atom_kernel<<<n_atoms, 32, 0, stream>>>(moments, (float*)d_out);
}